// CausalSequenceCML_21380347199617
// MI455X (gfx1250) — compile-verified
//
#include <hip/hip_runtime.h>

// Problem constants (match reference)
#define B_      4
#define T_      4096
#define C_      512
#define KW      4
#define STEPS_  16
#define CLAMP_  1.0e-4f

// Tiling: block = 512 threads = 16 waves (wave32).
// 32 channels per block (lane = channel  -> 128B coalesced access),
// 16 time-groups x 32 time-elems/thread = 512-point time region.
#define CB      32
#define TGP     16
#define RT      32
#define RTOT    (TGP * RT)            // 512 region length
#define HALO    (STEPS_ * (KW - 1))   // 48: contamination depth over 16 steps
#define TT      (RTOT - HALO)         // 464 valid outputs per block
#define NTB     ((T_ + TT - 1) / TT)  // 9 time blocks

// ---- CDNA5 async global->LDS staging (guarded; falls back to vec4 copy) ----
#if defined(__has_builtin)
#if __has_builtin(__builtin_amdgcn_global_load_async_to_lds_b128)
#define HAVE_ASYNC 1
#endif
#if __has_builtin(__builtin_amdgcn_s_wait_asynccnt)
#define HAVE_WAIT_ASYNC 1
#endif
#endif

typedef int v4i __attribute__((ext_vector_type(4)));
#if defined(HAVE_ASYNC)
typedef __attribute__((address_space(1))) v4i gv4i;   // global (AS1)
typedef __attribute__((address_space(3))) v4i lv4i;   // LDS (AS3)
#endif

__global__ __launch_bounds__(512) void cml_kernel(
    const float* __restrict__ drive, const float* __restrict__ r,
    const float* __restrict__ eps,   const float* __restrict__ beta,
    const float* __restrict__ Kc,    float* __restrict__ out)
{
    __shared__ float s_drv[RTOT * CB];            // [t][c], 64 KB
    __shared__ float s_halo[2][TGP][CB][3];       // double-buffered mapped tails

    const int tid     = threadIdx.x;
    const int c_local = tid & (CB - 1);
    const int tg      = tid >> 5;                 // wave index = time group
    const int b       = blockIdx.z;
    const int cb0     = blockIdx.y * CB;
    const int c       = cb0 + c_local;
    const int ts      = blockIdx.x * TT;          // first valid global t
    const int rt0     = ts - HALO;                // global t of region index 0

    // ---- stage drive tile (RTOT x CB f32) into LDS with 128-bit granules ----
    // tile has RTOT rows of 8 float4; vid = tid + k*512 ; t_loc = vid>>3 ; c4 = vid&7
    {
        #pragma unroll
        for (int k = 0; k < (RTOT * CB / 4) / 512; ++k) {
            int vid   = tid + k * 512;
            int t_loc = vid >> 3;
            int c4    = vid & 7;
            int gt    = rt0 + t_loc;
            gt = gt < 0 ? 0 : (gt > T_ - 1 ? T_ - 1 : gt);   // clamp: halo/tail values are sacrificial
            const float* gp = drive + ((size_t)(b * T_ + gt)) * C_ + cb0 + c4 * 4;
            float*       lp = &s_drv[t_loc * CB + c4 * 4];
#if defined(HAVE_ASYNC)
            __builtin_amdgcn_global_load_async_to_lds_b128(
                (gv4i*)(gp), (lv4i*)(lp), 0, 0);
#else
            *(float4*)lp = *(const float4*)gp;
#endif
        }
#if defined(HAVE_ASYNC)
#if defined(HAVE_WAIT_ASYNC)
        __builtin_amdgcn_s_wait_asynccnt(0);
#else
        asm volatile("s_wait_asynccnt 0" ::: "memory");
#endif
#endif
    }
    __syncthreads();

    // ---- per-channel parameters ----
    const float rc = r[c];
    const float ec = eps[c];
    const float bc = beta[c];
    const float w0 = Kc[c * KW + 0], w1 = Kc[c * KW + 1];
    const float w2 = Kc[c * KW + 2], w3 = Kc[c * KW + 3];
    const float ome = 1.0f - ec, omb = 1.0f - bc;

    const int tbase = tg * RT;                 // region-local strip start
    const int gt0   = rt0 + tbase;             // global t of strip start
    // number of leading strip elements with global t < 0 (zero-padded conv input)
    const int nneg  = gt0 >= 0 ? 0 : (-gt0 > RT ? RT : -gt0);

    float g[RT];
    #pragma unroll
    for (int j = 0; j < RT; ++j)
        g[j] = s_drv[(tbase + j) * CB + c_local];   // g0 = drive

    float m[RT];
    for (int s = 0; s < STEPS_; ++s) {
        // logistic map; force mapped=0 for global t<0 (matches zero padding)
        #pragma unroll
        for (int j = 0; j < RT; ++j) {
            float mm = rc * g[j] * (1.0f - g[j]);
            m[j] = (j >= nneg) ? mm : 0.0f;
        }
        // publish mapped tail (t0+RT-3 .. t0+RT-1) for the next time-group
        const int buf = s & 1;
        s_halo[buf][tg][c_local][0] = m[RT - 3];
        s_halo[buf][tg][c_local][1] = m[RT - 2];
        s_halo[buf][tg][c_local][2] = m[RT - 1];
        __syncthreads();                       // one barrier/step (double-buffered)

        float lm3 = 0.0f, lm2 = 0.0f, lm1 = 0.0f;   // mapped at t0-3, t0-2, t0-1
        if (tg > 0) {
            lm3 = s_halo[buf][tg - 1][c_local][0];
            lm2 = s_halo[buf][tg - 1][c_local][1];
            lm1 = s_halo[buf][tg - 1][c_local][2];
        }
        // causal 4-tap depthwise conv + relaxation + drive coupling
        #pragma unroll
        for (int j = 0; j < RT; ++j) {
            float a3 = (j >= 3) ? m[j - 3] : (j == 0 ? lm3 : (j == 1 ? lm2 : lm1));
            float a2 = (j >= 2) ? m[j - 2] : (j == 0 ? lm2 : lm1);
            float a1 = (j >= 1) ? m[j - 1] : lm1;
            float loc  = w0 * a3 + w1 * a2 + w2 * a1 + w3 * m[j];
            float phys = ome * m[j] + ec * loc;
            g[j] = omb * phys + bc * s_drv[(tbase + j) * CB + c_local];
        }
    }

    // ---- clamp + store the valid (non-halo) region, coalesced across c ----
    #pragma unroll
    for (int j = 0; j < RT; ++j) {
        int rl = tbase + j;
        int t  = rt0 + rl;
        if (rl >= HALO && t < T_) {
            float v = g[j];
            v = v < CLAMP_ ? CLAMP_ : v;
            v = v > 1.0f - CLAMP_ ? 1.0f - CLAMP_ : v;
            out[((size_t)(b * T_ + t)) * C_ + c] = v;
        }
    }
}

extern "C" void kernel_launch(void* const* d_in, const int* in_sizes, int n_in,
                              void* d_out, int out_size, void* d_ws, size_t ws_size,
                              hipStream_t stream) {
    (void)in_sizes; (void)n_in; (void)out_size; (void)d_ws; (void)ws_size;
    const float* drive = (const float*)d_in[0];
    const float* r     = (const float*)d_in[1];
    const float* eps   = (const float*)d_in[2];
    const float* beta  = (const float*)d_in[3];
    const float* Kc    = (const float*)d_in[4];
    // d_in[5] = steps (fixed 16, baked in as STEPS_)

    dim3 grid(NTB, C_ / CB, B_);
    cml_kernel<<<grid, 512, 0, stream>>>(drive, r, eps, beta, Kc, (float*)d_out);
}